// GCN2_Net_26912265077118
// MI455X (gfx1250) — compile-verified
//
#include <hip/hip_runtime.h>
#include <hip/hip_bf16.h>
#include <math.h>

typedef __attribute__((ext_vector_type(2))) float v2f;
typedef __attribute__((ext_vector_type(8))) float v8f;

#define D_IN   128
#define D_HID  64
#define D_OUT  64
#define NUM_LAYERS 8
#define ALPHA_ 0.1f
#define THETA_ 0.5f

// ---------------- degree / normalization ----------------

__global__ void init_deg_kernel(float* __restrict__ deg, int n) {
    int i = blockIdx.x * blockDim.x + threadIdx.x;
    if (i < n) deg[i] = 1.0f;   // self-loop contributes 1 to every node's degree
}

__global__ void accum_deg_kernel(const int* __restrict__ dst, float* __restrict__ deg, int e) {
    int i = blockIdx.x * blockDim.x + threadIdx.x;
    if (i < e) atomicAdd(&deg[dst[i]], 1.0f);
}

__global__ void dinv_kernel(const float* __restrict__ deg, float* __restrict__ dinv, int n) {
    int i = blockIdx.x * blockDim.x + threadIdx.x;
    if (i < n) {
        float d = deg[i];
        dinv[i] = (d > 0.0f) ? rsqrtf(d) : 0.0f;
    }
}

// ah[i,:] = dinv[i]^2 * h[i,:]   (self-loop term; also serves as the zero-init of ah)
__global__ void self_loop_kernel(const float* __restrict__ h, const float* __restrict__ dinv,
                                 float* __restrict__ ah, int n) {
    long long t = (long long)blockIdx.x * blockDim.x + threadIdx.x;
    int i = (int)(t >> 6);           // node
    if (i >= n) return;
    float di = dinv[i];
    ah[t] = di * di * h[t];
}

// ah[dst,:] += dinv[src]*dinv[dst] * h[src,:]   (64 threads per edge -> coalesced 128B per wave)
__global__ void edge_scatter_kernel(const int* __restrict__ src, const int* __restrict__ dst,
                                    const float* __restrict__ dinv, const float* __restrict__ h,
                                    float* __restrict__ ah, int e) {
    long long t = (long long)blockIdx.x * blockDim.x + threadIdx.x;
    int ed = (int)(t >> 6);
    if (ed >= e) return;
    int ch = (int)(t & 63);
    int s = src[ed];
    int d = dst[ed];
    float w = dinv[s] * dinv[d];
    atomicAdd(&ah[(long long)d * 64 + ch], w * h[(long long)s * 64 + ch]);
}

// ---------------- WMMA f32 GEMM, 16-row x 64-col tile per 128-thread block ----------------
//
//  xx  = ca*A0 + cb*A1               (FUSE: A1 != nullptr; else xx = A0)
//  out = act( cin*xx + cw*(xx @ W) [+ bias] )     W is [K x 64] row-major
//  IDENT (cin*xx) only meaningful for K == 64 (square layer).
//
// A-frag (16x4 f32, 2 VGPR/lane): lanes 0-15 -> K={k,k+1}, lanes 16-31 -> K={k+2,k+3}, M=lane%16
// B-frag (4x16 f32, 2 VGPR/lane): symmetric in K halves, N=lane%16
// C/D (16x16 f32, 8 VGPR): m = r + 8*(lane>=16), n = lane%16

template <int K, bool RELU, bool USE_BIAS, bool IDENT, bool FUSE>
__global__ __launch_bounds__(128)
void wmma_gemm16_kernel(const float* __restrict__ A0, const float* __restrict__ A1,
                        const float* __restrict__ W,  const float* __restrict__ bias,
                        float* __restrict__ out, float* __restrict__ out2,
                        int nrows, float ca, float cb, float cin, float cw) {
    __shared__ float As[16 * K];

    const int tid  = threadIdx.x;
    const int row0 = blockIdx.x * 16;

    // Stage the 16 x K A-tile into LDS (coalesced), fusing xx = ca*A0 + cb*A1.
    for (int idx = tid; idx < 16 * K; idx += 128) {
        int r = idx / K;
        int c = idx - r * K;
        int row = row0 + r;
        if (row >= nrows) row = nrows - 1;            // clamp (n is a multiple of 16 anyway)
        long long g = (long long)row * K + c;
        float v = A0[g];
        if (FUSE) v = ca * v + cb * A1[g];
        As[idx] = v;
    }
    __syncthreads();

    const int lane = tid & 31;
    const int wv   = tid >> 5;        // 0..3 : which 16-col output tile
    const int n0   = wv * 16;
    const int half = lane >> 4;       // 0 / 1
    const int l16  = lane & 15;

    v8f acc = {};
    #pragma unroll
    for (int k = 0; k < K; k += 4) {
        const int ka = k + 2 * half;
        v2f a, b;
        a.x = As[l16 * K + ka + 0];
        a.y = As[l16 * K + ka + 1];
        b.x = W[(ka + 0) * 64 + n0 + l16];
        b.y = W[(ka + 1) * 64 + n0 + l16];
        // D = A(16x4) * B(4x16) + C   -- v_wmma_f32_16x16x4_f32
        acc = __builtin_amdgcn_wmma_f32_16x16x4_f32(
            /*neg_a=*/false, a, /*neg_b=*/false, b,
            /*c_mod=*/(short)0, acc, /*reuse_a=*/false, /*reuse_b=*/false);
    }

    const float bv = USE_BIAS ? bias[n0 + l16] : 0.0f;

    #pragma unroll
    for (int r = 0; r < 8; ++r) {
        const int m = r + 8 * half;
        const int row = row0 + m;
        if (row >= nrows) continue;
        float v = cw * acc[r];
        if (IDENT)    v += cin * As[m * K + n0 + l16];   // (1-beta)*xx term (K==64)
        if (USE_BIAS) v += bv;
        if (RELU)     v = fmaxf(v, 0.0f);
        long long g = (long long)row * 64 + n0 + l16;
        out[g] = v;
        if (out2) out2[g] = v;
    }
}

// ---------------- host-side orchestration ----------------

extern "C" void kernel_launch(void* const* d_in, const int* in_sizes, int n_in,
                              void* d_out, int out_size, void* d_ws, size_t ws_size,
                              hipStream_t stream) {
    const float* x    = (const float*)d_in[0];   // [N, 128]
    const int*   edge = (const int*)  d_in[1];   // [2, E]  (row 0 = src, row 1 = dst)
    const float* W0   = (const float*)d_in[2];   // [128, 64]
    const float* b0   = (const float*)d_in[3];   // [64]
    const float* Ws   = (const float*)d_in[4];   // [8, 64, 64]
    const float* Wout = (const float*)d_in[5];   // [64, 64]
    const float* bout = (const float*)d_in[6];   // [64]
    float* out = (float*)d_out;

    const int n = in_sizes[0] / D_IN;            // 50000
    const int e = in_sizes[1] / 2;               // 800000
    const int* src = edge;
    const int* dst = edge + e;

    // Workspace carve-up (all f32): deg[n], dinv[n], h[n*64], h0[n*64], ah[n*64]
    char* ws = (char*)d_ws;
    float* deg  = (float*)ws;  ws += sizeof(float) * (size_t)n;
    float* dinv = (float*)ws;  ws += sizeof(float) * (size_t)n;
    float* h    = (float*)ws;  ws += sizeof(float) * (size_t)n * 64;
    float* h0   = (float*)ws;  ws += sizeof(float) * (size_t)n * 64;
    float* ah   = (float*)ws;  ws += sizeof(float) * (size_t)n * 64;

    const int TB = 256;
    const int nBlk  = (n + TB - 1) / TB;
    const int eBlk  = (e + TB - 1) / TB;
    const long long nc  = (long long)n * 64;
    const long long ec  = (long long)e * 64;
    const int ncBlk = (int)((nc + TB - 1) / TB);
    const int ecBlk = (int)((ec + TB - 1) / TB);
    const int gBlk  = (n + 15) / 16;             // 3125 tile-rows

    // 1) GCN normalization: deg (with self-loops) -> deg^-1/2
    init_deg_kernel<<<nBlk, TB, 0, stream>>>(deg, n);
    accum_deg_kernel<<<eBlk, TB, 0, stream>>>(dst, deg, e);
    dinv_kernel<<<nBlk, TB, 0, stream>>>(deg, dinv, n);

    // 2) h = relu(x @ W0 + b0); h0 = h
    wmma_gemm16_kernel<D_IN, true, true, false, false><<<gBlk, 128, 0, stream>>>(
        x, nullptr, W0, b0, h, h0, n, 1.0f, 0.0f, 0.0f, 1.0f);

    // 3) 8 GCNII layers
    for (int layer = 0; layer < NUM_LAYERS; ++layer) {
        float beta = logf(THETA_ / (float)(layer + 1) + 1.0f);

        self_loop_kernel<<<ncBlk, TB, 0, stream>>>(h, dinv, ah, n);
        edge_scatter_kernel<<<ecBlk, TB, 0, stream>>>(src, dst, dinv, h, ah, e);

        // xx = (1-alpha)*ah + alpha*h0 (fused in A-staging)
        // h  = relu( (1-beta)*xx + beta*(xx @ Ws[layer]) )
        wmma_gemm16_kernel<D_HID, true, false, true, true><<<gBlk, 128, 0, stream>>>(
            ah, h0, Ws + (size_t)layer * D_HID * D_HID, nullptr, h, nullptr,
            n, 1.0f - ALPHA_, ALPHA_, 1.0f - beta, beta);
    }

    // 4) out = h @ W_out + b_out
    wmma_gemm16_kernel<D_HID, false, true, false, false><<<gBlk, 128, 0, stream>>>(
        h, nullptr, Wout, bout, out, nullptr, n, 1.0f, 0.0f, 0.0f, 1.0f);
}